// By_Event_15977278341438
// MI455X (gfx1250) — compile-verified
//
#include <hip/hip_runtime.h>
#include <stdint.h>

#define LNUM     4096
#define EMAX     2048
#define NT       256
#define IOU_T    0.2f

// ---- LDS pool layout (bytes). stage(16KB) overlays MAXR+MAXC (used later). ----
#define OFF_MAXR   0        // f32[2048]   (pass1 row max, reused pass2)
#define OFF_MAXC   8192     // f32[2048]
#define OFF_EVOS   16384    // u16[2048]   output event starts (sorted)
#define OFF_EVOE   20480    // u16[2048]   output event ends   (sorted)
#define OFF_EVTS   24576    // u16[2048]   target event starts (sorted)
#define OFF_EVTE   28672    // u16[2048]
#define OFF_IDXR1  32768    // u16[2048]
#define OFF_IDXC1  36864    // u16[2048]
#define OFF_IDXR2  40960    // u16[2048]
#define OFF_IDXC2  45056    // u16[2048]
#define OFF_BIN    49152    // u8[4096]
#define OFF_RFLAG  53248    // u8[2048]  0=none 1=mutual 2=nonmutual-high
#define OFF_CFLAG  55296    // u8[2048]
#define OFF_SCANS  57344    // u32[256] (only 8 wave totals used now)
#define OFF_SCANE  58368    // u32[256]
#define OFF_MISC   59392    // u32[8]: [0]=extract count, [1]=tp1, [2]=tp2
#define POOL_BYTES 59424

__device__ __forceinline__ float iou_val(int as, int ae, int bs, int be) {
    int inter = min(ae, be) - max(as, bs);
    inter = max(inter, 0);
    int uni = (ae - as) + (be - bs) - inter;
    return (uni > 0) ? ((float)inter / (float)uni) : 0.0f;
}

// first index with arr[idx] > key (sorted ascending)
__device__ __forceinline__ int ub_u16(const uint16_t* arr, int n, int key) {
    int lo = 0, hi = n;
    while (lo < hi) { int mid = (lo + hi) >> 1; if ((int)arr[mid] > key) hi = mid; else lo = mid + 1; }
    return lo;
}
// first index with arr[idx] >= key
__device__ __forceinline__ int lb_u16(const uint16_t* arr, int n, int key) {
    int lo = 0, hi = n;
    while (lo < hi) { int mid = (lo + hi) >> 1; if ((int)arr[mid] >= key) hi = mid; else lo = mid + 1; }
    return lo;
}

// Extract run events from bin[0..L-1]; k-th start pairs with k-th end (alternation).
// Wave32 shuffle scans + 8-entry wave-total scan for the compaction offsets.
__device__ int extract_events(const uint8_t* bin, uint16_t* evs, uint16_t* eve,
                              uint32_t* scanS, uint32_t* scanE, uint32_t* misc) {
    const int t    = threadIdx.x;
    const int lane = t & 31;
    const int wave = t >> 5;
    const int i0 = t * 17;
    const int i1 = min(i0 + 17, LNUM + 1);   // boundary positions 0..L
    uint32_t cs = 0, ce = 0;
    for (int i = i0; i < i1; ++i) {
        uint32_t b  = (i < LNUM) ? bin[i]     : 0u;
        uint32_t bp = (i > 0)    ? bin[i - 1] : 0u;
        cs += (b & ~bp) & 1u;      // rising edge at i
        ce += (bp & ~b) & 1u;      // falling edge at i
    }
    // inclusive scan within wave32
    uint32_t sS = cs, sE = ce;
#pragma unroll
    for (int d = 1; d < 32; d <<= 1) {
        uint32_t uS = __shfl_up(sS, (unsigned)d, 32);
        uint32_t uE = __shfl_up(sE, (unsigned)d, 32);
        if (lane >= d) { sS += uS; sE += uE; }
    }
    if (lane == 31) { scanS[wave] = sS; scanE[wave] = sE; }   // wave totals
    __syncthreads();
    if (t == 0) {
        uint32_t aS = 0, aE = 0;
#pragma unroll
        for (int w = 0; w < NT / 32; ++w) {
            uint32_t x = scanS[w]; scanS[w] = aS; aS += x;
            uint32_t y = scanE[w]; scanE[w] = aE; aE += y;
        }
        misc[0] = aS;              // == aE
    }
    __syncthreads();
    uint32_t oS = scanS[wave] + (sS - cs);   // exclusive prefix for this thread
    uint32_t oE = scanE[wave] + (sE - ce);
    for (int i = i0; i < i1; ++i) {
        uint32_t b  = (i < LNUM) ? bin[i]     : 0u;
        uint32_t bp = (i > 0)    ? bin[i - 1] : 0u;
        if ((b & ~bp) & 1u) evs[oS++] = (uint16_t)i;
        if ((bp & ~b) & 1u) eve[oE++] = (uint16_t)i;
    }
    __syncthreads();
    return (int)misc[0];
}

// CDNA5 async global->LDS staging of 16KB row: 4x b128 per thread, ASYNCcnt path.
__device__ __forceinline__ void async_stage_row(const void* src_base, uint32_t lds_base_byte) {
    const int t = threadIdx.x;
    const char* g = (const char*)src_base + (size_t)t * 64;
    uint32_t l = lds_base_byte + (uint32_t)t * 64u;
#pragma unroll
    for (int k = 0; k < 4; ++k) {
        asm volatile("global_load_async_to_lds_b128 %0, %1, off"
                     :: "v"(l + (uint32_t)(k * 16)), "v"(g + k * 16) : "memory");
    }
    asm volatile("s_wait_asynccnt 0" ::: "memory");
}

__global__ void __launch_bounds__(NT) init_out_kernel(float* out) {
    if (threadIdx.x < 3) out[threadIdx.x] = 0.0f;
}

__global__ void __launch_bounds__(NT) event_metrics_kernel(
    const float* __restrict__ prob, const int* __restrict__ tgt, float* __restrict__ out)
{
    __shared__ __align__(16) unsigned char pool[POOL_BYTES];
    float*    maxr  = (float*)   (pool + OFF_MAXR);
    float*    maxc  = (float*)   (pool + OFF_MAXC);
    uint16_t* evos  = (uint16_t*)(pool + OFF_EVOS);
    uint16_t* evoe  = (uint16_t*)(pool + OFF_EVOE);
    uint16_t* evts  = (uint16_t*)(pool + OFF_EVTS);
    uint16_t* evte  = (uint16_t*)(pool + OFF_EVTE);
    uint16_t* idxr1 = (uint16_t*)(pool + OFF_IDXR1);
    uint16_t* idxc1 = (uint16_t*)(pool + OFF_IDXC1);
    uint16_t* idxr2 = (uint16_t*)(pool + OFF_IDXR2);
    uint16_t* idxc2 = (uint16_t*)(pool + OFF_IDXC2);
    uint8_t*  bin   = (uint8_t*) (pool + OFF_BIN);
    uint8_t*  rflag = (uint8_t*) (pool + OFF_RFLAG);
    uint8_t*  cflag = (uint8_t*) (pool + OFF_CFLAG);
    uint32_t* scanS = (uint32_t*)(pool + OFF_SCANS);
    uint32_t* scanE = (uint32_t*)(pool + OFF_SCANE);
    uint32_t* misc  = (uint32_t*)(pool + OFF_MISC);
    float*    stage_f = (float*)(pool + OFF_MAXR);   // 16KB staging aliases maxr+maxc
    const int*  stage_i = (const int*)stage_f;

    const int row = blockIdx.x;
    const int t   = threadIdx.x;
    const uint32_t stage_lds = (uint32_t)(uintptr_t)stage_f;  // low 32 bits of flat LDS addr

    // ---- stage prob row via async DMA, binarize at 0.5 ----
    async_stage_row(prob + (size_t)row * LNUM, stage_lds);
    __syncthreads();
    for (int k = 0; k < 16; ++k) {
        int i = t * 16 + k;
        bin[i] = (stage_f[i] >= 0.5f) ? 1u : 0u;
    }
    __syncthreads();

    // ---- extract, remove length-1 events, re-extract ----
    int n1 = extract_events(bin, evos, evoe, scanS, scanE, misc);
    for (int k = t; k < n1; k += NT) {
        if ((int)evoe[k] - (int)evos[k] == 1) bin[evos[k]] = 0;  // dur==1 -> zeroed
    }
    __syncthreads();
    int n_out = extract_events(bin, evos, evoe, scanS, scanE, misc);

    // ---- stage target row via async DMA, binarize (!=0), extract ----
    async_stage_row(tgt + (size_t)row * LNUM, stage_lds);
    __syncthreads();
    for (int k = 0; k < 16; ++k) {
        int i = t * 16 + k;
        bin[i] = (stage_i[i] != 0) ? 1u : 0u;
    }
    __syncthreads();
    int n_tgt = extract_events(bin, evts, evte, scanS, scanE, misc);

    // ---- pass 1: row/col max + first-occurrence argmax of IoU (windowed) ----
    // Events are sorted & disjoint: overlaps of [as,ae) are exactly the contiguous
    // range [first be > as, first bs >= ae). Out-of-window IoU == 0, in-window > 0,
    // so windowed scan with best=0/idx=0 init replicates jnp.argmax exactly.
    for (int i = t; i < n_out; i += NT) {
        int as = evos[i], ae = evoe[i];
        float best = 0.0f; int bj = 0;
        int jlo  = ub_u16(evte, n_tgt, as);
        int jend = lb_u16(evts, n_tgt, ae);
        for (int j = jlo; j < jend; ++j) {
            float v = iou_val(as, ae, evts[j], evte[j]);
            if (v > best) { best = v; bj = j; }
        }
        maxr[i] = best; idxr1[i] = (uint16_t)bj;
    }
    for (int j = t; j < n_tgt; j += NT) {
        int bs = evts[j], be = evte[j];
        float best = 0.0f; int bi = 0;
        int ilo  = ub_u16(evoe, n_out, bs);
        int iend = lb_u16(evos, n_out, be);
        for (int i = ilo; i < iend; ++i) {
            float v = iou_val(evos[i], evoe[i], bs, be);
            if (v > best) { best = v; bi = i; }
        }
        maxc[j] = best; idxc1[j] = (uint16_t)bi;
    }
    if (t == 0) { misc[1] = 0u; misc[2] = 0u; }
    __syncthreads();

    // ---- pass 1 flags + tp1 ----
    {
        uint32_t local = 0;
        for (int i = t; i < n_out; i += NT) {
            uint8_t f = 0;
            if (maxr[i] >= IOU_T) {
                f = ((int)idxc1[idxr1[i]] == i) ? 1u : 2u;
                if (f == 1u) local++;
            }
            rflag[i] = f;
        }
        if (local) atomicAdd(&misc[1], local);
        for (int j = t; j < n_tgt; j += NT) {
            uint8_t f = 0;
            if (maxc[j] >= IOU_T) f = ((int)idxr1[idxc1[j]] == j) ? 1u : 2u;
            cflag[j] = f;
        }
    }
    __syncthreads();

    // ---- pass 2: same matching on bm1-masked IoU (col_mut == mut_tgt identity).
    // Every bm1 cell has IoU >= 0.2 > 0, hence lies inside the overlap window.
    for (int i = t; i < n_out; i += NT) {
        int as = evos[i], ae = evoe[i];
        bool i_nm = (rflag[i] == 2u), i_notmut = (rflag[i] != 1u);
        int jr = idxr1[i];
        float best = 0.0f; int bj = 0;
        int jlo  = ub_u16(evte, n_tgt, as);
        int jend = lb_u16(evts, n_tgt, ae);
        for (int j = jlo; j < jend; ++j) {
            bool brow = i_nm && (jr == j) && (cflag[j] != 1u);
            bool bcol = (cflag[j] == 2u) && ((int)idxc1[j] == i) && i_notmut;
            if (brow || bcol) {
                float v = iou_val(as, ae, evts[j], evte[j]);
                if (v > best) { best = v; bj = j; }
            }
        }
        maxr[i] = best; idxr2[i] = (uint16_t)bj;
    }
    for (int j = t; j < n_tgt; j += NT) {
        int bs = evts[j], be = evte[j];
        bool j_nm = (cflag[j] == 2u), j_notmut = (cflag[j] != 1u);
        int ic = idxc1[j];
        float best = 0.0f; int bi = 0;
        int ilo  = ub_u16(evoe, n_out, bs);
        int iend = lb_u16(evos, n_out, be);
        for (int i = ilo; i < iend; ++i) {
            bool brow = (rflag[i] == 2u) && ((int)idxr1[i] == j) && j_notmut;
            bool bcol = j_nm && (ic == i) && (rflag[i] != 1u);
            if (brow || bcol) {
                float v = iou_val(evos[i], evoe[i], bs, be);
                if (v > best) { best = v; bi = i; }
            }
        }
        maxc[j] = best; idxc2[j] = (uint16_t)bi;
    }
    __syncthreads();

    // ---- tp2: mutual best of masked matrix ----
    {
        uint32_t local = 0;
        for (int i = t; i < n_out; i += NT) {
            if (maxr[i] >= IOU_T && (int)idxc2[idxr2[i]] == i) local++;
        }
        if (local) atomicAdd(&misc[2], local);
    }
    __syncthreads();

    if (t == 0) {
        int tp = (int)(misc[1] + misc[2]);
        atomicAdd(out + 0, (float)tp);
        atomicAdd(out + 1, (float)(n_tgt - tp));
        atomicAdd(out + 2, (float)(n_out - tp));
    }
}

extern "C" void kernel_launch(void* const* d_in, const int* in_sizes, int n_in,
                              void* d_out, int out_size, void* d_ws, size_t ws_size,
                              hipStream_t stream) {
    (void)n_in; (void)out_size; (void)d_ws; (void)ws_size;
    const float* prob = (const float*)d_in[0];
    const int*   tgt  = (const int*)d_in[1];
    float*       out  = (float*)d_out;
    const int B = in_sizes[0] / LNUM;
    init_out_kernel<<<1, NT, 0, stream>>>(out);
    event_metrics_kernel<<<B, NT, 0, stream>>>(prob, tgt, out);
}